// DifferentiableCounter_13022340842131
// MI455X (gfx1250) — compile-verified
//
#include <hip/hip_runtime.h>

// DifferentiableCounter soft-counter scan for gfx1250 (MI455X).
// d_{t+1} = A_t d_t with A_t tridiagonal(64) + wrap; outputs pre-update states.
//
// Phase 1: per-chunk transition matrices, column-per-thread register-resident FMA.
// Phase 2: Hillis-Steele prefix scan of 64x64 matrices via V_WMMA_F32_16X16X4_F32.
// Phase 3: per-chunk emit; start state = column 0 of inclusive prefix (init = e0).
//
// Workspace requirement: 2 * NCHUNK * 4096 * 4 bytes = 32 MiB.

#define SEQ_LEN   (1 << 20)
#define NSTATE    64
#define NCHUNK    1024
#define CHUNK_LEN (SEQ_LEN / NCHUNK)     // 1024
#define MAT_ELEMS (NSTATE * NSTATE)      // 4096
#define COEF_TILE 128

typedef float v2f __attribute__((ext_vector_type(2)));
typedef float v8f __attribute__((ext_vector_type(8)));

// ---------------- Phase 1: chunk transition matrices ----------------
// Block = 64 threads (2 waves). Thread j owns column j of M_c, fully in VGPRs.
__global__ __launch_bounds__(64)
void k_chunk_matrices(const float* __restrict__ inc_p,
                      const float* __restrict__ dec_p,
                      float* __restrict__ M) {
  const int c = blockIdx.x;
  const int j = threadIdx.x;            // column index 0..63

  float m[NSTATE];
  #pragma unroll
  for (int k = 0; k < NSTATE; ++k) m[k] = (k == j) ? 1.0f : 0.0f;

  __shared__ float coef[2 * COEF_TILE];
  const int t0 = c * CHUNK_LEN;

  for (int base = 0; base < CHUNK_LEN; base += COEF_TILE) {
    for (int s = j; s < COEF_TILE; s += 64) {
      coef[2 * s]     = inc_p[t0 + base + s];
      coef[2 * s + 1] = dec_p[t0 + base + s];
    }
    __syncthreads();
    for (int s = 0; s < COEF_TILE; ++s) {
      const float inc  = coef[2 * s];
      const float dec  = coef[2 * s + 1];
      const float noop = fmaxf(0.0f, 1.0f - inc - dec);
      // In-place apply A_t to this column (m' = A_t m).
      const float old63 = m[63];
      float prev = m[0];
      m[0] = fmaf(inc, old63, fmaf(dec, prev + m[1], noop * prev));
      #pragma unroll
      for (int k = 1; k <= 62; ++k) {
        const float cur = m[k];
        m[k] = fmaf(inc, prev, fmaf(dec, m[k + 1], noop * cur));
        prev = cur;
      }
      m[63] = fmaf(inc, prev, noop * old63);
    }
    __syncthreads();
  }

  float* out = M + (size_t)c * MAT_ELEMS + j;   // row-major M[c][k][j]
  #pragma unroll
  for (int k = 0; k < NSTATE; ++k) out[k * NSTATE] = m[k];
}

// ---------------- Phase 2: WMMA matrix prefix scan (Hillis-Steele step) -----
// dst[c] = src[c] * src[c-hop]  (compose newer-on-left; column-vector states)
// Block = 256 threads (8 waves); 4x4 grid of 16x16 C tiles, 2 tiles per wave.
__global__ __launch_bounds__(256)
void k_scan_step(const float* __restrict__ src,
                 float* __restrict__ dst, int hop) {
  const int c   = blockIdx.x;
  const int tid = threadIdx.x;

  if (c < hop) {  // block-uniform branch: plain copy
    const float4* s = (const float4*)(src + (size_t)c * MAT_ELEMS);
    float4*       d = (float4*)(dst + (size_t)c * MAT_ELEMS);
    for (int i = tid; i < MAT_ELEMS / 4; i += 256) d[i] = s[i];
    return;
  }

  __shared__ float Lm[MAT_ELEMS];  // left matrix, row-major
  __shared__ float Rt[MAT_ELEMS];  // right matrix, transposed (Rt[n][k] = R[k][n])
  {
    const float* L = src + (size_t)c * MAT_ELEMS;
    const float* R = src + (size_t)(c - hop) * MAT_ELEMS;
    for (int i = tid; i < MAT_ELEMS; i += 256) {
      Lm[i] = L[i];
      const int r = i >> 6, cc = i & 63;
      Rt[cc * NSTATE + r] = R[i];
    }
  }
  __syncthreads();

  const int lane = tid & 31;
  const int wave = tid >> 5;                // 0..7
  const int mn   = lane & 15;               // A row / B col within tile
  const int koff = (lane >> 4) * 2;         // f32 16x16x4 fragment K offset

  float* D = dst + (size_t)c * MAT_ELEMS;

  #pragma unroll
  for (int tt = 0; tt < 2; ++tt) {
    const int tile = wave * 2 + tt;         // 0..15
    const int ti = tile >> 2, tj = tile & 3;
    v8f acc = {};
    #pragma unroll
    for (int kk = 0; kk < 16; ++kk) {       // K = 64 in steps of 4
      const v2f a = *(const v2f*)&Lm[(ti * 16 + mn) * NSTATE + kk * 4 + koff];
      const v2f b = *(const v2f*)&Rt[(tj * 16 + mn) * NSTATE + kk * 4 + koff];
      acc = __builtin_amdgcn_wmma_f32_16x16x4_f32(
          false, a, false, b, (short)0, acc, false, false);
    }
    const int rowbase = ti * 16 + ((lane >> 4) << 3);  // lanes 16-31 hold M=r+8
    const int col     = tj * 16 + (lane & 15);
    #pragma unroll
    for (int r = 0; r < 8; ++r)
      D[(rowbase + r) * NSTATE + col] = acc[r];
  }
}

// ---------------- Phase 3: emit per-step pre-update states ----------------
// One wave per chunk. Lane j holds d[2j], d[2j+1]; neighbors via wave32 shuffles.
__global__ __launch_bounds__(32)
void k_emit(const float* __restrict__ inc_p,
            const float* __restrict__ dec_p,
            const float* __restrict__ P,   // inclusive prefix matrices
            float* __restrict__ out) {
  const int c = blockIdx.x;
  const int j = threadIdx.x;               // 0..31

  float lo, hi;
  if (c == 0) {
    lo = (j == 0) ? 1.0f : 0.0f;           // init dist = e0
    hi = 0.0f;
  } else {
    const float* Pm = P + (size_t)(c - 1) * MAT_ELEMS;
    lo = Pm[(2 * j)     * NSTATE];         // column 0 of prefix = start state
    hi = Pm[(2 * j + 1) * NSTATE];
  }

  const int t0 = c * CHUNK_LEN;
  for (int base = 0; base < CHUNK_LEN; base += 32) {
    const float my_inc = inc_p[t0 + base + j];
    const float my_dec = dec_p[t0 + base + j];
    for (int s = 0; s < 32; ++s) {
      const float inc  = __shfl(my_inc, s, 32);
      const float dec  = __shfl(my_dec, s, 32);
      const float noop = fmaxf(0.0f, 1.0f - inc - dec);

      // write PRE-update state (coalesced 256B per wave per step)
      const int t = t0 + base + s;
      float2 o; o.x = lo; o.y = hi;
      *(float2*)&out[(size_t)t * NSTATE + 2 * j] = o;

      const float up_lo = __shfl(hi, (j + 31) & 31, 32); // d[2j-1], wraps to d[63]
      const float nx_lo = __shfl(lo, (j + 1) & 31, 32);  // d[2j+2]
      const float dn_lo = hi + ((j == 0) ? lo : 0.0f);   // d[2j+1] (+ d[0] stays)
      const float dn_hi = (j == 31) ? 0.0f : nx_lo;      // d[2j+2], top clipped

      const float nlo = fmaf(inc, up_lo, fmaf(dec, dn_lo, noop * lo));
      const float nhi = fmaf(inc, lo,    fmaf(dec, dn_hi, noop * hi));
      lo = nlo; hi = nhi;
    }
  }
}

// ---------------- Host-side orchestration (all on `stream`) ----------------
extern "C" void kernel_launch(void* const* d_in, const int* in_sizes, int n_in,
                              void* d_out, int out_size, void* d_ws, size_t ws_size,
                              hipStream_t stream) {
  (void)in_sizes; (void)n_in; (void)out_size; (void)ws_size;
  const float* inc_p = (const float*)d_in[0];
  const float* dec_p = (const float*)d_in[1];
  float* out = (float*)d_out;

  float* P0 = (float*)d_ws;                          // NCHUNK matrices
  float* P1 = P0 + (size_t)NCHUNK * MAT_ELEMS;       // ping-pong buffer

  k_chunk_matrices<<<NCHUNK, 64, 0, stream>>>(inc_p, dec_p, P0);

  float* src = P0;
  float* dst = P1;
  for (int hop = 1; hop < NCHUNK; hop <<= 1) {       // 10 WMMA scan steps
    k_scan_step<<<NCHUNK, 256, 0, stream>>>(src, dst, hop);
    float* tmp = src; src = dst; dst = tmp;
  }
  // 10 swaps: inclusive prefixes end up back in P0 (== src).

  k_emit<<<NCHUNK, 32, 0, stream>>>(inc_p, dec_p, src, out);
}